// Conv2d_NN_spatial_7559142441291
// MI455X (gfx1250) — compile-verified
//
#include <hip/hip_runtime.h>

#define NB   16
#define NC   64
#define NH   96
#define NW   96
#define NN   (NH * NW)      // 9216 tokens
#define NS   64             // sample candidates
#define NK   3
#define TOKB 128            // tokens per block (8 waves x 16)
#define WAVES 8

typedef __attribute__((ext_vector_type(2))) float v2f;
typedef __attribute__((ext_vector_type(8))) float v8f;

// ---------------------------------------------------------------------------
// Phase 1 (one block per batch):
//   wsS [b][c][s]  = sample matrix (64x64)
//   wsS2[b][s]     = sum_c S^2
//   wsP [b][k][co][j] = sum_ci W[co][ci][k] * x[b][ci][j] , j in [0,64)
// ---------------------------------------------------------------------------
__global__ __launch_bounds__(256) void prep_kernel(
    const float* __restrict__ x, const float* __restrict__ w,
    float* __restrict__ wsS, float* __restrict__ wsS2, float* __restrict__ wsP)
{
    __shared__ float sG[NC * 64];   // first 64 tokens, [c][j]
    __shared__ float sS[NC * 64];   // sample matrix,   [c][s]
    const int b   = blockIdx.x;
    const int tid = threadIdx.x;
    const float* xb = x + (size_t)b * NC * NN;
    const int XI[8] = {0, 14, 27, 41, 54, 68, 81, 95}; // round(linspace(0,95,8))

    for (int t = tid; t < NC * 64; t += 256) {
        int c = t >> 6, j = t & 63;
        sG[t] = xb[c * NN + j];
        float sv = xb[c * NN + XI[j >> 3] * NW + XI[j & 7]];
        sS[t] = sv;
        wsS[(size_t)b * NC * 64 + t] = sv;
    }
    __syncthreads();

    if (tid < 64) {
        float acc = 0.f;
        for (int c = 0; c < NC; ++c) { float v = sS[c * 64 + tid]; acc += v * v; }
        wsS2[b * 64 + tid] = acc;
    }

    // P[k][co][j], idx = k*4096 + co*64 + j
    for (int idx = tid; idx < NK * NC * 64; idx += 256) {
        int k   = idx >> 12;
        int rem = idx & 4095;
        int co  = rem >> 6, j = rem & 63;
        float acc = 0.f;
        #pragma unroll 8
        for (int ci = 0; ci < NC; ++ci)
            acc += w[co * (NC * NK) + ci * NK + k] * sG[ci * 64 + j];
        wsP[(size_t)b * NK * NC * 64 + idx] = acc;
    }
}

// ---------------------------------------------------------------------------
// Phase 2: per wave: 16 tokens, WMMA f32 GEMM vs 64 candidates,
// top-3 smallest key (stable), gather-add from P, ReLU, store.
// ---------------------------------------------------------------------------
__global__ __launch_bounds__(256) void main_kernel(
    const float* __restrict__ x, const float* __restrict__ bias,
    const float* __restrict__ wsS, const float* __restrict__ wsS2,
    const float* __restrict__ wsP, float* __restrict__ out)
{
    __shared__ float sS[NC * 64];                 // 16 KB  sample matrix [c][s]
    __shared__ float sS2[64];
    __shared__ float sBias[64];
    __shared__ float keybuf[WAVES * 16 * 64];     // 32 KB
    __shared__ int   sIdx[TOKB * 3];

    const int b    = blockIdx.y;
    const int tid  = threadIdx.x;
    const int lane = tid & 31;
    const int wav  = tid >> 5;
    const int m16  = lane & 15;
    const int khalf = lane >> 4;                  // 0: lanes 0-15, 1: lanes 16-31

    const float* xb = x + (size_t)b * NC * NN;

    for (int t = tid; t < NC * 64; t += 256)
        sS[t] = wsS[(size_t)b * NC * 64 + t];
    if (tid < 64) {
        sS2[tid]  = wsS2[b * 64 + tid];
        sBias[tid] = bias[tid];
    }
    __syncthreads();

    const int n0 = blockIdx.x * TOKB + wav * 16;

    // dot[m][s] for m in [n0, n0+16), s in [0,64): 4 D-tiles, K-loop over C=64.
    v8f acc0 = {}, acc1 = {}, acc2 = {}, acc3 = {};
    const float* aptr = xb + n0 + m16;
    for (int kk = 0; kk < NC; kk += 4) {
        const int kA = kk + khalf * 2;
        v2f a;
        a.x = aptr[(size_t)kA * NN];
        a.y = aptr[(size_t)(kA + 1) * NN];
        const float* br0 = &sS[kA * 64 + m16];
        const float* br1 = &sS[(kA + 1) * 64 + m16];
        v2f b0; b0.x = br0[0];  b0.y = br1[0];
        v2f b1; b1.x = br0[16]; b1.y = br1[16];
        v2f b2; b2.x = br0[32]; b2.y = br1[32];
        v2f b3; b3.x = br0[48]; b3.y = br1[48];
        acc0 = __builtin_amdgcn_wmma_f32_16x16x4_f32(false, a, false, b0, (short)0, acc0, false, false);
        acc1 = __builtin_amdgcn_wmma_f32_16x16x4_f32(false, a, false, b1, (short)0, acc1, false, false);
        acc2 = __builtin_amdgcn_wmma_f32_16x16x4_f32(false, a, false, b2, (short)0, acc2, false, false);
        acc3 = __builtin_amdgcn_wmma_f32_16x16x4_f32(false, a, false, b3, (short)0, acc3, false, false);
    }

    // key[m][s] = s2[s] - 2*dot[m][s]  (ordering-equivalent to sqrt(n2+s2-2dot))
    float* kb = &keybuf[wav * 16 * 64];
    #pragma unroll
    for (int t = 0; t < 4; ++t) {
        const int s = t * 16 + m16;
        const float s2v = sS2[s];
        v8f c = (t == 0) ? acc0 : (t == 1) ? acc1 : (t == 2) ? acc2 : acc3;
        #pragma unroll
        for (int r = 0; r < 8; ++r) {
            const int M = r + 8 * khalf;            // C/D layout: VGPR r -> M=r (+8 hi lanes)
            kb[M * 64 + s] = s2v - 2.0f * c[r];
        }
    }
    __syncthreads();

    // top-3 smallest, stable (earlier index wins ties) -> matches lax.top_k
    if (lane < 16) {
        const float* row = kb + lane * 64;
        float v0 = 3.4e38f, v1 = 3.4e38f, v2 = 3.4e38f;
        int i0 = 0, i1 = 0, i2 = 0;
        #pragma unroll 4
        for (int s = 0; s < 64; ++s) {
            float v = row[s];
            if (v < v0)      { v2 = v1; i2 = i1; v1 = v0; i1 = i0; v0 = v; i0 = s; }
            else if (v < v1) { v2 = v1; i2 = i1; v1 = v;  i1 = s; }
            else if (v < v2) { v2 = v;  i2 = s; }
        }
        const int tk = wav * 16 + lane;
        sIdx[tk * 3 + 0] = i0; sIdx[tk * 3 + 1] = i1; sIdx[tk * 3 + 2] = i2;
    }
    __syncthreads();

    // out[b][co][n0+m] = ReLU(bias[co] + P0[co][i0] + P1[co][i1] + P2[co][i2])
    {
        const int tk = wav * 16 + m16;
        const int i0 = sIdx[tk * 3 + 0];
        const int i1 = sIdx[tk * 3 + 1];
        const int i2 = sIdx[tk * 3 + 2];
        const float* P  = wsP + (size_t)b * NK * NC * 64;
        const float* P0 = P + i0;
        const float* P1 = P + 4096 + i1;
        const float* P2 = P + 8192 + i2;
        float* ob = out + (size_t)b * NC * NN + n0 + m16;
        #pragma unroll 4
        for (int it = 0; it < 32; ++it) {
            const int co = it * 2 + khalf;
            float v = sBias[co] + P0[co * 64] + P1[co * 64] + P2[co * 64];
            ob[(size_t)co * NN] = fmaxf(v, 0.0f);
        }
    }
}

extern "C" void kernel_launch(void* const* d_in, const int* in_sizes, int n_in,
                              void* d_out, int out_size, void* d_ws, size_t ws_size,
                              hipStream_t stream)
{
    const float* x    = (const float*)d_in[0];   // [16,64,96,96]
    const float* w    = (const float*)d_in[1];   // [64,64,3]
    const float* bias = (const float*)d_in[2];   // [64]
    float* out = (float*)d_out;                  // [16,64,96,96]

    float* ws   = (float*)d_ws;
    float* wsS  = ws;                                   // 16*4096
    float* wsS2 = ws + (size_t)NB * NC * 64;            // 16*64
    float* wsP  = wsS2 + (size_t)NB * 64;               // 16*3*4096

    prep_kernel<<<dim3(NB), dim3(256), 0, stream>>>(x, w, wsS, wsS2, wsP);
    main_kernel<<<dim3(NN / TOKB, NB), dim3(256), 0, stream>>>(x, bias, wsS, wsS2, wsP, out);
}